// PoissonBivector_68289980007047
// MI455X (gfx1250) — compile-verified
//
#include <hip/hip_runtime.h>
#include <hip/hip_bf16.h>

typedef __attribute__((ext_vector_type(2))) float v2f;
typedef __attribute__((ext_vector_type(8))) float v8f;

namespace {

constexpr int kDim  = 6;
constexpr int kTile = 16;                       // samples per wave (N columns of WMMA tile)
constexpr int kWaves = 8;                       // waves per 256-thread block
constexpr int kBlockSamples = kTile * kWaves;   // 128

// row-major triu(k=1) index pairs for dim=6 (matches jnp.triu_indices order)
__device__ constexpr int IU[15] = {0,0,0,0,0,1,1,1,1,2,2,2,3,3,4};
__device__ constexpr int JU[15] = {1,2,3,4,5,2,3,4,5,3,4,5,4,5,5};

__device__ __forceinline__ float sp_act(float v) {
  // stable softplus: max(v,0) + ln(1 + exp(-|v|)), using raw v_exp_f32/v_log_f32
  // (base-2 HW transcendentals; args never denormal: exp2 in <= 0, log2 in (1,2])
  float e  = __builtin_amdgcn_exp2f(-fabsf(v) * 1.4426950408889634f);
  float l2 = __builtin_amdgcn_logf(1.0f + e);
  return fmaf(0.6931471805599453f, l2, fmaxf(v, 0.0f));
}

__device__ __forceinline__ v8f wmma_k8(v2f a0, v2f a1, v2f b0, v2f b1, v8f c) {
  // D = A(16x8,f32) x B(8x16,f32) + C via two chained 16x16x4 fp32 WMMAs
  c = __builtin_amdgcn_wmma_f32_16x16x4_f32(false, a0, false, b0, (short)0, c, false, false);
  c = __builtin_amdgcn_wmma_f32_16x16x4_f32(false, a1, false, b1, (short)0, c, false, false);
  return c;
}

// A-tile loader for an (nrows x 8) row-major weight matrix, zero-padded to 16 rows.
// Unconditional load from a clamped row + cndmask (no exec divergence).
__device__ __forceinline__ void load_A8(const float* W, int m, int half, int nrows,
                                        v2f& a0, v2f& a1) {
  const v2f zero2 = {0.0f, 0.0f};
  int mm = (m < nrows) ? m : 0;
  const float* wr = W + mm * 8 + 2 * half;
  v2f t0 = *(const v2f*)(wr);
  v2f t1 = *(const v2f*)(wr + 4);
  bool z = (m >= nrows);
  a0 = z ? zero2 : t0;
  a1 = z ? zero2 : t1;
}

// Bias C-init for hidden layers (8 valid rows): rows 8..15 are zero.
// All loads unconditional at compile-time offsets -> scalar loads + cndmask.
__device__ __forceinline__ v8f bias_c8(const float* b, int half) {
  v8f c;
#pragma unroll
  for (int j = 0; j < 8; ++j) {
    float t = b[j];
    c[j] = half ? 0.0f : t;
  }
  return c;
}

// Bias C-init for the 15-row output layer: row m = j + 8*half, row 15 -> 0.
__device__ __forceinline__ v8f bias_c15(const float* b, int half) {
  v8f c;
#pragma unroll
  for (int j = 0; j < 8; ++j) {
    float lo = b[j];                            // rows 0..7
    float hi = (j < 7) ? b[8 + j] : 0.0f;       // rows 8..14, row 15 = 0 (compile-time)
    c[j] = half ? hi : lo;
  }
  return c;
}

// softplus on accumulator rows 0..7, then rebuild the next-layer B tile:
// B vgpr j must hold feature (j + 2*half) of sample idx; features live in lane idx.
__device__ __forceinline__ void next_b(const v8f& acc, int half, int src,
                                       v2f& b0, v2f& b1) {
  float y[8];
#pragma unroll
  for (int j = 0; j < 8; ++j) y[j] = sp_act(acc[j]);
  float t2 = __shfl(y[2], src, 32);
  float t3 = __shfl(y[3], src, 32);
  float t6 = __shfl(y[6], src, 32);
  float t7 = __shfl(y[7], src, 32);
  b0.x = half ? t2 : y[0];
  b0.y = half ? t3 : y[1];
  b1.x = half ? t6 : y[4];
  b1.y = half ? t7 : y[5];
}

__global__ __launch_bounds__(256) void PoissonBivector_68289980007047_kernel(
    const float* __restrict__ x,
    const float* __restrict__ Win, const float* __restrict__ bin,
    const float* __restrict__ W1, const float* __restrict__ bv1,
    const float* __restrict__ W2, const float* __restrict__ bv2,
    const float* __restrict__ W3, const float* __restrict__ bv3,
    const float* __restrict__ W4, const float* __restrict__ bv4,
    const float* __restrict__ Wout, const float* __restrict__ bout,
    float* __restrict__ out, long long N) {
  __shared__ __align__(16) float Lbuf[kWaves * kTile * 36];  // 18432 B

  const int lane = (int)(threadIdx.x & 31);
  const int warp = (int)(threadIdx.x >> 5);
  const int idx  = lane & 15;   // sample-in-tile / matrix row index
  const int half = lane >> 4;   // lane group (selects K/M upper halves)

  const long long tileBase = (long long)blockIdx.x * kBlockSamples + (long long)warp * kTile;
  if (tileBase >= N) return;    // wave-uniform

  long long s = tileBase + idx;
  if (s >= N) s = N - 1;        // clamp for partial tail tile (stores guarded later)

  const v2f zero2 = {0.0f, 0.0f};

  // ---- input B tile: B[k=feature][n=sample], features 0..5 (6,7 zero-padded)
  const float* xs = x + s * kDim;
  v2f b0 = *(const v2f*)(xs + 2 * half);     // features {2h, 2h+1}
  v2f b1t = *(const v2f*)(xs + 4);           // features {4,5} (valid memory for all lanes)
  v2f b1 = half ? zero2 : b1t;               // half=1 -> features {6,7} = 0

  const int m = idx;
  v2f a0, a1;

  // ---- layer in: W_in is 8x6 row-major
  {
    int mm = (m < 8) ? m : 0;
    const float* wr = Win + mm * kDim;
    v2f t0 = *(const v2f*)(wr + 2 * half);   // K = {2h, 2h+1}
    v2f t1 = *(const v2f*)(wr + 4);          // K = {4,5}
    bool z = (m >= 8);
    a0 = z ? zero2 : t0;
    a1 = (z || half) ? zero2 : t1;           // K = {6,7} zero
  }
  v8f acc = wmma_k8(a0, a1, b0, b1, bias_c8(bin, half));

  // ---- 4 hidden layers (8x8)
  const float* Ws[4] = {W1, W2, W3, W4};
  const float* bs[4] = {bv1, bv2, bv3, bv4};
#pragma unroll
  for (int L = 0; L < 4; ++L) {
    next_b(acc, half, idx, b0, b1);
    load_A8(Ws[L], m, half, 8, a0, a1);
    acc = wmma_k8(a0, a1, b0, b1, bias_c8(bs[L], half));
  }

  // ---- output layer: W_out is 15x8
  next_b(acc, half, idx, b0, b1);
  load_A8(Wout, m, half, 15, a0, a1);
  acc = wmma_k8(a0, a1, b0, b1, bias_c15(bout, half));
  // acc vgpr j = ut[m = j + 8*half][sample idx]  (row 15 invalid, ignored)

  // ---- stage antisymmetric 6x6 matrices in LDS (compile-time triu offsets)
  float* Ls = &Lbuf[warp * (kTile * 36) + idx * 36];
  if (!half) {
#pragma unroll
    for (int k = 0; k < 8; ++k) {
      float v = acc[k];
      Ls[IU[k] * 6 + JU[k]] = v;
      Ls[JU[k] * 6 + IU[k]] = -v;
    }
    Ls[0] = 0.0f; Ls[7] = 0.0f; Ls[14] = 0.0f;
    Ls[21] = 0.0f; Ls[28] = 0.0f; Ls[35] = 0.0f;
  } else {
#pragma unroll
    for (int j = 0; j < 7; ++j) {
      int k = 8 + j;
      float v = acc[j];
      Ls[IU[k] * 6 + JU[k]] = v;
      Ls[JU[k] * 6 + IU[k]] = -v;
    }
  }

  // wave-local DS ops complete in order; wait, then stream out coalesced 8B stores
  asm volatile("s_wait_dscnt 0x0" ::: "memory");

  const v2f* s2 = (const v2f*)&Lbuf[warp * (kTile * 36)];
  v2f* d2 = (v2f*)(out + tileBase * 36);
  if (tileBase + kTile <= N) {
    // full tile: unguarded coalesced stream (common case)
#pragma unroll
    for (int r = 0; r < 9; ++r) {         // 288 float2 per wave = 9 iters x 32 lanes
      int t = r * 32 + lane;
      d2[t] = s2[t];
    }
  } else {
    const long long lim2 = N * 18;        // total float2 in output
    const long long base2 = tileBase * 18;
#pragma unroll
    for (int r = 0; r < 9; ++r) {
      int t = r * 32 + lane;
      if (base2 + t < lim2) d2[t] = s2[t];
    }
  }
}

}  // namespace

extern "C" void kernel_launch(void* const* d_in, const int* in_sizes, int n_in,
                              void* d_out, int out_size, void* d_ws, size_t ws_size,
                              hipStream_t stream) {
  const float* x    = (const float*)d_in[0];
  const float* Win  = (const float*)d_in[1];
  const float* bin  = (const float*)d_in[2];
  const float* W1   = (const float*)d_in[3];
  const float* bv1  = (const float*)d_in[4];
  const float* W2   = (const float*)d_in[5];
  const float* bv2  = (const float*)d_in[6];
  const float* W3   = (const float*)d_in[7];
  const float* bv3  = (const float*)d_in[8];
  const float* W4   = (const float*)d_in[9];
  const float* bv4  = (const float*)d_in[10];
  const float* Wout = (const float*)d_in[11];
  const float* bout = (const float*)d_in[12];
  float* out = (float*)d_out;

  long long N = (long long)in_sizes[0] / kDim;  // 1,048,576 for the reference shapes
  int blocks = (int)((N + kBlockSamples - 1) / kBlockSamples);

  PoissonBivector_68289980007047_kernel<<<blocks, 256, 0, stream>>>(
      x, Win, bin, W1, bv1, W2, bv2, W3, bv3, W4, bv4, Wout, bout, out, N);
}